// DGCNN_76836964926096
// MI455X (gfx1250) — compile-verified
//
#include <hip/hip_runtime.h>
#include <hip/hip_bf16.h>

// ---------------------------------------------------------------------------
// DGCNN forward for MI455X (gfx1250, wave32, WMMA).
// Edge conv factorized: y[b,n,k,o] = P[b,o,idx] - P[b,o,n] + Q[b,o,n]
// with P = W[:, :C] @ x, Q = W[:, C:] @ x  (dense WMMA GEMMs).
// Channel counts are template constants -> branchless unrolled WMMA K-loops.
// LDS staging uses async global->LDS copies (ASYNCcnt) when available.
// ---------------------------------------------------------------------------

typedef __attribute__((ext_vector_type(2))) float v2f;
typedef __attribute__((ext_vector_type(8))) float v8f;

#define BATCH 8
#define NPTS  2048
#define KNN   20

#if defined(__has_builtin)
#  if __has_builtin(__builtin_amdgcn_global_load_async_to_lds_b32) && \
      __has_builtin(__builtin_amdgcn_s_wait_asynccnt)
#    define HAVE_ASYNC_LDS 1
#  endif
#endif

#ifdef HAVE_ASYNC_LDS
// builtin params are int* in AS(1) (global) / AS(3) (LDS)
#define GAS1(p) ((__attribute__((address_space(1))) int*)(p))
#define LAS3(p) ((__attribute__((address_space(3))) int*)(p))
// async copy of one dword: global -> LDS, tracked by ASYNCcnt
#define ASYNC_CP32(gp, lp) \
    __builtin_amdgcn_global_load_async_to_lds_b32(GAS1(gp), LAS3(lp), 0, 0)
#define ASYNC_WAIT() __builtin_amdgcn_s_wait_asynccnt(0)
#endif

__device__ __forceinline__ v8f wmma4(v2f a, v2f b, v8f c) {
    // D(16x16,f32) = A(16x4,f32) * B(4x16,f32) + C
    return __builtin_amdgcn_wmma_f32_16x16x4_f32(
        false, a, false, b, (short)0, c, false, false);
}

// -------------------- squared norms: xx[b][n] = sum_c x[b][c][n]^2 ----------
__global__ void dg_sqnorm(const float* __restrict__ x, float* __restrict__ xx,
                          int C, long xBatchStride) {
    int b = blockIdx.y;
    int n = blockIdx.x * blockDim.x + threadIdx.x;
    const float* xb = x + (long)b * xBatchStride;
    float s = 0.f;
    for (int c = 0; c < C; ++c) {
        float t = xb[(long)c * NPTS + n];
        s += t * t;
    }
    xx[(long)b * NPTS + n] = s;
}

// -------------------- Gram: gram[b][q][d] = sum_c x[c][q]*x[c][d] -----------
// Block: 256 threads = 8 waves. Each wave: one 16x16 tile (q-tile per wave,
// shared d-tile). x[0:C][d0:d0+16] staged in LDS, reused by all 8 waves.
template<int C>
__global__ void __launch_bounds__(256) dg_gram(const float* __restrict__ x,
                                               float* __restrict__ gram,
                                               long xBatchStride) {
    constexpr int NC = (C + 3) & ~3;
    const float* xb = x + (long)blockIdx.z * xBatchStride;
    float* gb = gram + (long)blockIdx.z * NPTS * NPTS;
    __shared__ float lB[C][17];                 // padded: conflict-free
    int t  = threadIdx.x;
    int d0 = blockIdx.y * 16;
    for (int e = t; e < C * 16; e += 256) {
        int k = e >> 4, j = e & 15;
#ifdef HAVE_ASYNC_LDS
        ASYNC_CP32(xb + (long)k * NPTS + d0 + j, &lB[k][j]);
#else
        lB[k][j] = xb[(long)k * NPTS + d0 + j]; // 16-wide coalesced rows
#endif
    }
#ifdef HAVE_ASYNC_LDS
    ASYNC_WAIT();
#endif
    __syncthreads();

    int wid = t >> 5, lane = t & 31;
    int half = lane >> 4, lm = lane & 15;
    int q = blockIdx.x * 128 + wid * 16 + lm;   // A-row index for loads
    v8f acc = {0.f,0.f,0.f,0.f,0.f,0.f,0.f,0.f};
#pragma unroll
    for (int c0 = 0; c0 < NC; c0 += 4) {
        int k0 = c0 + half * 2;
        int ka = (k0     < C) ? k0     : 0;     // branchless tail (C=3 only)
        int kb = (k0 + 1 < C) ? k0 + 1 : 0;
        v2f a, bv;
        float a0 = xb[(long)ka * NPTS + q];
        float a1 = xb[(long)kb * NPTS + q];
        a.x  = (k0     < C) ? a0 : 0.f;
        a.y  = (k0 + 1 < C) ? a1 : 0.f;
        bv.x = (k0     < C) ? lB[ka][lm] : 0.f;
        bv.y = (k0 + 1 < C) ? lB[kb][lm] : 0.f;
        acc = wmma4(a, bv, acc);
    }
    int n     = d0 + lm;
    int mbase = blockIdx.x * 128 + wid * 16 + half * 8;
#pragma unroll
    for (int v = 0; v < 8; ++v)
        gb[(long)(mbase + v) * NPTS + n] = acc[v];
}

// -------------------- top-20 select, LDS-tiled candidate scan ---------------
// 256 queries per block; candidates staged 32 at a time with fully coalesced
// 128B bursts (async->LDS when available); per-thread register insertion sort
// (stable in m -> matches lax.top_k tie-breaking).
// Rank key: 2*inner - xx[m]   (-xx[n] is constant per query).
__global__ void __launch_bounds__(256) dg_topk(const float* __restrict__ gram,
                                               const float* __restrict__ xx,
                                               int* __restrict__ idx) {
    constexpr int TM = 32;
    __shared__ float tile[256][TM + 1];
    __shared__ float xs[TM];
    int b  = blockIdx.y;
    int t  = threadIdx.x;
    int n0 = blockIdx.x * 256;
    const float* gb  = gram + (long)b * NPTS * NPTS;
    const float* xxb = xx + (long)b * NPTS;
    float v[KNN]; int id[KNN];
#pragma unroll
    for (int j = 0; j < KNN; ++j) { v[j] = -3.4e38f; id[j] = 0; }

    int wid = t >> 5, lane = t & 31;
    for (int m0 = 0; m0 < NPTS; m0 += TM) {
        __syncthreads();                        // previous tile fully consumed
#pragma unroll
        for (int r = 0; r < 32; ++r) {
            int q = r * 8 + wid;                // 8 rows per pass, 32 lanes wide
#ifdef HAVE_ASYNC_LDS
            ASYNC_CP32(gb + (long)(n0 + q) * NPTS + m0 + lane, &tile[q][lane]);
#else
            tile[q][lane] = gb[(long)(n0 + q) * NPTS + m0 + lane];
#endif
        }
        if (t < TM) xs[t] = xxb[m0 + t];
#ifdef HAVE_ASYNC_LDS
        ASYNC_WAIT();
#endif
        __syncthreads();
        for (int mm = 0; mm < TM; ++mm) {
            float val = 2.0f * tile[t][mm] - xs[mm];
            if (val > v[KNN - 1]) {
                float cv = val; int ci = m0 + mm;
#pragma unroll
                for (int j = 0; j < KNN; ++j) {
                    if (cv > v[j]) {
                        float tv = v[j]; int ti = id[j];
                        v[j] = cv; id[j] = ci;
                        cv = tv; ci = ti;
                    }
                }
            }
        }
    }
    int* irow = idx + ((long)b * NPTS + n0 + t) * KNN;
#pragma unroll
    for (int j = 0; j < KNN; ++j) irow[j] = id[j];
}

// -------------------- P/Q GEMM: P[o][n]=W[o][0:C].x ; Q uses W[o][C:2C] -----
// Block: 256 threads = 8 waves sharing one 16-o W-tile staged in LDS
// (padded rows), each wave a different 16-n tile.
template<int C>
__global__ void __launch_bounds__(256) dg_pq(const float* __restrict__ x,
                                             const float* __restrict__ W,
                                             float* __restrict__ P,
                                             float* __restrict__ Q,
                                             int O, long xBatchStride) {
    constexpr int twoC = 2 * C;
    constexpr int NC   = (C + 3) & ~3;
    constexpr int LW   = twoC + 4;              // pad -> conflict-free A reads
    __shared__ float lA[16][LW];
    int b = blockIdx.z;
    const float* xb = x + (long)b * xBatchStride;
    float* Pb = P + (long)b * O * NPTS;
    float* Qb = Q + (long)b * O * NPTS;
    int t  = threadIdx.x;
    int o0 = blockIdx.x * 16;
    for (int e = t; e < 16 * twoC; e += 256) {
        int r = e / twoC, c = e % twoC;
#ifdef HAVE_ASYNC_LDS
        ASYNC_CP32(W + (long)(o0 + r) * twoC + c, &lA[r][c]);
#else
        lA[r][c] = W[(long)(o0 + r) * twoC + c];
#endif
    }
#ifdef HAVE_ASYNC_LDS
    ASYNC_WAIT();
#endif
    __syncthreads();

    int wid = t >> 5, lane = t & 31;
    int half = lane >> 4, lm = lane & 15;
    int n = blockIdx.y * 128 + wid * 16 + lm;
    v8f accP = {0.f,0.f,0.f,0.f,0.f,0.f,0.f,0.f};
    v8f accQ = {0.f,0.f,0.f,0.f,0.f,0.f,0.f,0.f};
#pragma unroll
    for (int c0 = 0; c0 < NC; c0 += 4) {
        int k0 = c0 + half * 2;
        int ka = (k0     < C) ? k0     : 0;
        int kb = (k0 + 1 < C) ? k0 + 1 : 0;
        v2f bv, aP, aQ;
        float x0 = xb[(long)ka * NPTS + n];
        float x1 = xb[(long)kb * NPTS + n];
        bv.x = (k0     < C) ? x0 : 0.f;
        bv.y = (k0 + 1 < C) ? x1 : 0.f;
        aP.x = (k0     < C) ? lA[lm][k0]         : 0.f;
        aP.y = (k0 + 1 < C) ? lA[lm][k0 + 1]     : 0.f;
        aQ.x = (k0     < C) ? lA[lm][C + k0]     : 0.f;
        aQ.y = (k0 + 1 < C) ? lA[lm][C + k0 + 1] : 0.f;
        accP = wmma4(aP, bv, accP);
        accQ = wmma4(aQ, bv, accQ);
    }
    int nn = blockIdx.y * 128 + wid * 16 + lm;
    int ob = o0 + half * 8;
#pragma unroll
    for (int v = 0; v < 8; ++v) {
        Pb[(long)(ob + v) * NPTS + nn] = accP[v];
        Qb[(long)(ob + v) * NPTS + nn] = accQ[v];
    }
}

// -------------------- BN stats over (b,n,k) per o: sum & sumsq --------------
__global__ void dg_ec_stats(const float* __restrict__ P, const float* __restrict__ Q,
                            const int* __restrict__ idx, float* __restrict__ partial,
                            int O) {
    int b = blockIdx.z, o = blockIdx.y;
    int n = blockIdx.x * blockDim.x + threadIdx.x;
    const float* Po = P + ((long)b * O + o) * NPTS;
    const float* Qo = Q + ((long)b * O + o) * NPTS;
    const int* irow = idx + ((long)b * NPTS + n) * KNN;
    float base = Qo[n] - Po[n];
    float s = 0.f, s2 = 0.f;
#pragma unroll
    for (int k = 0; k < KNN; ++k) {
        float val = Po[irow[k]] + base;
        s  += val;
        s2 += val * val;
    }
    __shared__ float sh[256], sh2[256];
    int t = threadIdx.x;
    sh[t] = s; sh2[t] = s2;
    __syncthreads();
    for (int st = 128; st > 0; st >>= 1) {
        if (t < st) { sh[t] += sh[t + st]; sh2[t] += sh2[t + st]; }
        __syncthreads();
    }
    if (t == 0) {
        long pi = (((long)b * O + o) * gridDim.x + blockIdx.x) * 2;
        partial[pi]     = sh[0];
        partial[pi + 1] = sh2[0];
    }
}

// -------------------- reduce partials -> per-channel scale/shift ------------
__global__ void dg_bn_reduce(const float* __restrict__ partial,
                             const float* __restrict__ g, const float* __restrict__ beta,
                             float* __restrict__ scsh,
                             int O, int nblk, float invCount) {
    int o = blockIdx.x * blockDim.x + threadIdx.x;
    if (o >= O) return;
    float s = 0.f, s2 = 0.f;
    for (int b = 0; b < BATCH; ++b)
        for (int p = 0; p < nblk; ++p) {
            long pi = (((long)b * O + o) * nblk + p) * 2;
            s  += partial[pi];
            s2 += partial[pi + 1];
        }
    float mean  = s * invCount;
    float var   = s2 * invCount - mean * mean;
    float scale = g[o] * rsqrtf(var + 1e-5f);
    float shift = beta[o] - mean * scale;
    scsh[o * 2]     = scale;
    scsh[o * 2 + 1] = shift;
}

// -------------------- BN + leaky + max over k -> xcat channel slice ---------
__global__ void dg_ec_maxout(const float* __restrict__ P, const float* __restrict__ Q,
                             const int* __restrict__ idx, const float* __restrict__ scsh,
                             float* __restrict__ xout, int O, long outBatchStride) {
    int b = blockIdx.z, o = blockIdx.y;
    int n = blockIdx.x * blockDim.x + threadIdx.x;
    const float* Po = P + ((long)b * O + o) * NPTS;
    const float* Qo = Q + ((long)b * O + o) * NPTS;
    const int* irow = idx + ((long)b * NPTS + n) * KNN;
    float base  = Qo[n] - Po[n];
    float scale = scsh[o * 2];
    float shift = scsh[o * 2 + 1];
    float m = -3.4e38f;
#pragma unroll
    for (int k = 0; k < KNN; ++k) {
        float val = (Po[irow[k]] + base) * scale + shift;
        val = val > 0.f ? val : 0.2f * val;
        m = fmaxf(m, val);
    }
    xout[(long)b * outBatchStride + (long)o * NPTS + n] = m;
}

// -------------------- final GEMM: Y[b][o][n] = sum_c W5[o][c]*xcat[b][c][n] -
// Kc = 512 fixed. 16-o W5 tile (32KB+pad) staged in LDS, 8 waves x 16 n.
__global__ void __launch_bounds__(256) dg_gemm5(const float* __restrict__ W,
                                                const float* __restrict__ X,
                                                float* __restrict__ Y) {
    constexpr int Kc = 512;
    constexpr int LW = Kc + 4;
    __shared__ float lA[16][LW];
    int b = blockIdx.z;
    const float* Xb = X + (long)b * Kc * NPTS;
    float* Yb = Y + (long)b * 1024 * NPTS;
    int t  = threadIdx.x;
    int o0 = blockIdx.x * 16;
    for (int e = t; e < 16 * Kc; e += 256) {
        int r = e >> 9, c = e & 511;
#ifdef HAVE_ASYNC_LDS
        ASYNC_CP32(W + ((long)(o0 + r) << 9) + c, &lA[r][c]);
#else
        lA[r][c] = W[((long)(o0 + r) << 9) + c];
#endif
    }
#ifdef HAVE_ASYNC_LDS
    ASYNC_WAIT();
#endif
    __syncthreads();

    int wid = t >> 5, lane = t & 31;
    int half = lane >> 4, lm = lane & 15;
    int n = blockIdx.y * 128 + wid * 16 + lm;
    v8f acc = {0.f,0.f,0.f,0.f,0.f,0.f,0.f,0.f};
#pragma unroll 8
    for (int c0 = 0; c0 < Kc; c0 += 4) {
        int k0 = c0 + half * 2;
        v2f a, bv;
        a.x  = lA[lm][k0];
        a.y  = lA[lm][k0 + 1];
        bv.x = Xb[(long)(k0    ) * NPTS + n];
        bv.y = Xb[(long)(k0 + 1) * NPTS + n];
        acc = wmma4(a, bv, acc);
    }
    int ob = o0 + half * 8;
#pragma unroll
    for (int v = 0; v < 8; ++v)
        Yb[(long)(ob + v) * NPTS + n] = acc[v];
}

// -------------------- final-layer BN stats over (b,n) per o -----------------
__global__ void dg_y_stats(const float* __restrict__ Y, float* __restrict__ partial,
                           int O) {
    int b = blockIdx.z, o = blockIdx.y;
    int n = blockIdx.x * blockDim.x + threadIdx.x;
    float val = Y[((long)b * O + o) * NPTS + n];
    __shared__ float sh[256], sh2[256];
    int t = threadIdx.x;
    sh[t] = val; sh2[t] = val * val;
    __syncthreads();
    for (int st = 128; st > 0; st >>= 1) {
        if (t < st) { sh[t] += sh[t + st]; sh2[t] += sh2[t + st]; }
        __syncthreads();
    }
    if (t == 0) {
        long pi = (((long)b * O + o) * gridDim.x + blockIdx.x) * 2;
        partial[pi]     = sh[0];
        partial[pi + 1] = sh2[0];
    }
}

// -------------------- BN + leaky + max/mean pool -> out [B, 2048] -----------
__global__ void dg_final_out(const float* __restrict__ Y, const float* __restrict__ scsh,
                             float* __restrict__ out, int O) {
    int b = blockIdx.y, o = blockIdx.x;
    const float* Yo = Y + ((long)b * O + o) * NPTS;
    float scale = scsh[o * 2];
    float shift = scsh[o * 2 + 1];
    float m = -3.4e38f, s = 0.f;
    for (int n = threadIdx.x; n < NPTS; n += blockDim.x) {
        __builtin_prefetch(Yo + n + 256, 0, 0);   // global_prefetch_b8
        float v = Yo[n] * scale + shift;
        v = v > 0.f ? v : 0.2f * v;
        m = fmaxf(m, v);
        s += v;
    }
    __shared__ float shm[256], shs[256];
    int t = threadIdx.x;
    shm[t] = m; shs[t] = s;
    __syncthreads();
    for (int st = 128; st > 0; st >>= 1) {
        if (t < st) {
            shm[t] = fmaxf(shm[t], shm[t + st]);
            shs[t] += shs[t + st];
        }
        __syncthreads();
    }
    if (t == 0) {
        out[(long)b * (2 * O) + o]     = shm[0];
        out[(long)b * (2 * O) + O + o] = shs[0] / (float)NPTS;
    }
}

// ---------------------------------------------------------------------------
extern "C" void kernel_launch(void* const* d_in, const int* in_sizes, int n_in,
                              void* d_out, int out_size, void* d_ws, size_t ws_size,
                              hipStream_t stream) {
    const float* x  = (const float*)d_in[0];
    const float* Wl[4] = { (const float*)d_in[1], (const float*)d_in[2],
                           (const float*)d_in[3], (const float*)d_in[4] };
    const float* W5 = (const float*)d_in[5];
    const float* gl[4] = { (const float*)d_in[6],  (const float*)d_in[8],
                           (const float*)d_in[10], (const float*)d_in[12] };
    const float* bl[4] = { (const float*)d_in[7],  (const float*)d_in[9],
                           (const float*)d_in[11], (const float*)d_in[13] };
    const float* g5 = (const float*)d_in[14];
    const float* b5 = (const float*)d_in[15];
    float* out = (float*)d_out;

    // workspace layout (floats)
    float* ws      = (float*)d_ws;
    float* xcat    = ws;                                   // 8*512*2048
    float* gram    = xcat + 8L * 512 * NPTS;               // 8*2048*2048 (reused as ybuf)
    float* ybuf    = gram;                                 // 8*1024*2048 fits inside gram
    float* xx      = gram + 8L * NPTS * NPTS;              // 8*2048
    int*   idxb    = (int*)(xx + 8L * NPTS);               // 8*2048*20
    float* P       = (float*)(idxb + 8L * NPTS * KNN);     // 8*256*2048
    float* Q       = P + 8L * 256 * NPTS;                  // 8*256*2048
    float* partial = Q + 8L * 256 * NPTS;                  // 8*1024*8*2
    float* scsh    = partial + 8L * 1024 * 8 * 2;          // 1024*2

    const int Cin[4]     = { 3, 64, 64, 128 };
    const int Oc[4]      = { 64, 64, 128, 256 };
    const int coffIn[4]  = { 0, 0, 64, 128 };
    const int coffOut[4] = { 0, 64, 128, 256 };
    const int nblk = NPTS / 256;                // 8 stat blocks per (b, o)

    for (int li = 0; li < 4; ++li) {
        const int C = Cin[li], O = Oc[li];
        const float* xin;
        long xstride;
        if (li == 0) { xin = x; xstride = 3L * NPTS; }
        else         { xin = xcat + (long)coffIn[li] * NPTS; xstride = 512L * NPTS; }

        dim3 gGram(NPTS / 128, NPTS / 16, BATCH);
        dim3 gPq(O / 16, NPTS / 128, BATCH);

        dg_sqnorm<<<dim3(NPTS / 256, BATCH), 256, 0, stream>>>(xin, xx, C, xstride);
        if (li == 0) {
            dg_gram<3><<<gGram, 256, 0, stream>>>(xin, gram, xstride);
        } else if (C == 64) {
            dg_gram<64><<<gGram, 256, 0, stream>>>(xin, gram, xstride);
        } else {
            dg_gram<128><<<gGram, 256, 0, stream>>>(xin, gram, xstride);
        }
        dg_topk<<<dim3(NPTS / 256, BATCH), 256, 0, stream>>>(gram, xx, idxb);
        if (li == 0) {
            dg_pq<3><<<gPq, 256, 0, stream>>>(xin, Wl[li], P, Q, O, xstride);
        } else if (C == 64) {
            dg_pq<64><<<gPq, 256, 0, stream>>>(xin, Wl[li], P, Q, O, xstride);
        } else {
            dg_pq<128><<<gPq, 256, 0, stream>>>(xin, Wl[li], P, Q, O, xstride);
        }
        dg_ec_stats<<<dim3(nblk, O, BATCH), 256, 0, stream>>>(P, Q, idxb, partial, O);
        dg_bn_reduce<<<dim3((O + 255) / 256), 256, 0, stream>>>(
            partial, gl[li], bl[li], scsh, O, nblk,
            1.0f / (float)(BATCH * NPTS * KNN));
        dg_ec_maxout<<<dim3(nblk, O, BATCH), 256, 0, stream>>>(
            P, Q, idxb, scsh, xcat + (long)coffOut[li] * NPTS, O, 512L * NPTS);
    }

    // final 512 -> 1024 conv + BN + leaky + max/mean pool
    dg_gemm5<<<dim3(1024 / 16, NPTS / 128, BATCH), 256, 0, stream>>>(W5, xcat, ybuf);
    dg_y_stats<<<dim3(nblk, 1024, BATCH), 256, 0, stream>>>(ybuf, partial, 1024);
    dg_bn_reduce<<<dim3(4), 256, 0, stream>>>(partial, g5, b5, scsh, 1024, nblk,
                                              1.0f / (float)(BATCH * NPTS));
    dg_final_out<<<dim3(1024, BATCH), 256, 0, stream>>>(ybuf, scsh, out, 1024);
}